// KNNTM_42649025249987
// MI455X (gfx1250) — compile-verified
//
#include <hip/hip_runtime.h>
#include <math.h>

#define V_DIM 5000
#define N_DIM 10000
#define T_DIM 50
#define H_DIM 200
#define B_DIM 1024
#define K_SEL 20
#define ALPHA_C 1.0f
#define ETA_C 0.5f
#define RHO_C 0.5f
#define EPS_BN 1e-5f
#define CC_C 0.25f

typedef __attribute__((ext_vector_type(2))) float v2f;
typedef __attribute__((ext_vector_type(8))) float v8f;

__device__ __forceinline__ float softplusf(float x) {
  return fmaxf(x, 0.f) + log1pf(expf(-fabsf(x)));
}

__device__ __forceinline__ v8f wmma_f32(v2f a, v2f b, v8f c) {
  return __builtin_amdgcn_wmma_f32_16x16x4_f32(false, a, false, b, (short)0, c, false, false);
}

// ---------------------------------------------------------------------------
// WMMA f32 GEMM:  out[M,N] = act(A[M,K] @ W[N,K]^T + bias)
// One wave per 16x16 tile. Guards fully hoisted out of the K loop:
//  - A rows always valid (M multiple of 16)
//  - B column index clamped to N-1: those lanes compute garbage for columns
//    that are never stored (store has the col<N guard), so no masking needed.
// Main loop: K&~7 with two accumulators (breaks WMMA->WMMA RAW chain),
// then K&~3, then one guarded tail step (only for K=50).
// ---------------------------------------------------------------------------
__global__ __launch_bounds__(32) void wmma_gemm_nt(
    const float* __restrict__ A, const float* __restrict__ W,
    const float* __restrict__ bias, float* __restrict__ out,
    int M, int N, int K, int act)
{
  const int m0   = blockIdx.x * 16;
  const int n0   = blockIdx.y * 16;
  const int lane = threadIdx.x;
  const int r    = lane & 15;
  const int h    = lane >> 4;
  const int rowA = m0 + r;
  const int colN = n0 + r;
  const int colW = (colN < N) ? colN : (N - 1);   // clamp, never stored if OOB
  const float* Ap = A + (size_t)rowA * K + 2*h;   // lane-biased base pointers
  const float* Wp = W + (size_t)colW * K + 2*h;

  v8f c0 = {0.f,0.f,0.f,0.f,0.f,0.f,0.f,0.f};
  v8f c1 = {0.f,0.f,0.f,0.f,0.f,0.f,0.f,0.f};
  const int K8 = K & ~7;
  const int K4 = K & ~3;
  int k = 0;
  for (; k < K8; k += 8) {
    const float2 a0 = *reinterpret_cast<const float2*>(Ap + k);
    const float2 b0 = *reinterpret_cast<const float2*>(Wp + k);
    const float2 a1 = *reinterpret_cast<const float2*>(Ap + k + 4);
    const float2 b1 = *reinterpret_cast<const float2*>(Wp + k + 4);
    const v2f av0 = {a0.x, a0.y}, bv0 = {b0.x, b0.y};
    const v2f av1 = {a1.x, a1.y}, bv1 = {b1.x, b1.y};
    c0 = wmma_f32(av0, bv0, c0);
    c1 = wmma_f32(av1, bv1, c1);
  }
  for (; k < K4; k += 4) {
    const float2 a0 = *reinterpret_cast<const float2*>(Ap + k);
    const float2 b0 = *reinterpret_cast<const float2*>(Wp + k);
    const v2f av = {a0.x, a0.y}, bv = {b0.x, b0.y};
    c0 = wmma_f32(av, bv, c0);
  }
  if (k < K) {                       // K%4 != 0 tail (K=50 -> lanes h=1 get 0)
    const int kk = k + 2*h;
    v2f a = {0.f, 0.f}, b = {0.f, 0.f};
    if (kk + 1 < K) {
      const float2 ta = *reinterpret_cast<const float2*>(Ap + k);
      a.x = ta.x; a.y = ta.y;
      const float2 tb = *reinterpret_cast<const float2*>(Wp + k);
      b.x = tb.x; b.y = tb.y;
    } else if (kk < K) {
      a.x = Ap[k]; b.x = Wp[k];
    }
    c0 = wmma_f32(a, b, c0);
  }
  #pragma unroll
  for (int i = 0; i < 8; ++i) c0[i] += c1[i];

  if (colN < N) {
    #pragma unroll
    for (int i = 0; i < 8; ++i) {
      const int row = m0 + i + 8*h;
      float v = c0[i];
      if (bias) v += bias[colN];
      if (act == 1) v = softplusf(v);
      out[(size_t)row * N + colN] = v;
    }
  }
}

// ---------------------------------------------------------------------------
// Cost GEMM + fused fuse-matrix epilogue (N = 10000 = 625*16, no col tail):
//   dot = soft[B,T] @ theta_bank[N,T]^T ; cost = s2[b] + tb2[n] - 2*dot
//   fuse = ETA*cost^2 + (1-ETA)*(RHO*Mcos[idx[b],n] + (1-RHO)*Mcoo[idx[b],n])
//   fuse[b, idx[b]] = +inf
// ---------------------------------------------------------------------------
__global__ __launch_bounds__(32) void wmma_cost(
    const float* __restrict__ soft, const float* __restrict__ tbank,
    const float* __restrict__ s2v, const float* __restrict__ tb2v,
    const int* __restrict__ idx,
    const float* __restrict__ Mcos, const float* __restrict__ Mcoo,
    float* __restrict__ fuse)
{
  const int m0   = blockIdx.x * 16;
  const int n0   = blockIdx.y * 16;
  const int lane = threadIdx.x;
  const int r    = lane & 15;
  const int h    = lane >> 4;
  const float* Ap = soft  + (size_t)(m0 + r) * T_DIM + 2*h;
  const float* Wp = tbank + (size_t)(n0 + r) * T_DIM + 2*h;

  v8f c0 = {0.f,0.f,0.f,0.f,0.f,0.f,0.f,0.f};
  v8f c1 = {0.f,0.f,0.f,0.f,0.f,0.f,0.f,0.f};
  const int K8 = T_DIM & ~7;   // 48
  int k = 0;
  for (; k < K8; k += 8) {
    const float2 a0 = *reinterpret_cast<const float2*>(Ap + k);
    const float2 b0 = *reinterpret_cast<const float2*>(Wp + k);
    const float2 a1 = *reinterpret_cast<const float2*>(Ap + k + 4);
    const float2 b1 = *reinterpret_cast<const float2*>(Wp + k + 4);
    const v2f av0 = {a0.x, a0.y}, bv0 = {b0.x, b0.y};
    const v2f av1 = {a1.x, a1.y}, bv1 = {b1.x, b1.y};
    c0 = wmma_f32(av0, bv0, c0);
    c1 = wmma_f32(av1, bv1, c1);
  }
  {                                   // tail: k = 48, kk = 48|50
    const int kk = k + 2*h;
    v2f a = {0.f, 0.f}, b = {0.f, 0.f};
    if (kk + 1 < T_DIM) {
      const float2 ta = *reinterpret_cast<const float2*>(Ap + k);
      a.x = ta.x; a.y = ta.y;
      const float2 tb = *reinterpret_cast<const float2*>(Wp + k);
      b.x = tb.x; b.y = tb.y;
    }
    c0 = wmma_f32(a, b, c0);
  }
  #pragma unroll
  for (int i = 0; i < 8; ++i) c0[i] += c1[i];

  const int n = n0 + r;
  #pragma unroll
  for (int i = 0; i < 8; ++i) {
    const int brow = m0 + i + 8*h;
    const float dot  = c0[i];
    const float cost = s2v[brow] + tb2v[n] - 2.f*dot;
    const float th   = cost * cost;
    const int   ib   = idx[brow];
    const float bow  = RHO_C * Mcos[(size_t)ib*N_DIM + n]
                     + (1.f-RHO_C) * Mcoo[(size_t)ib*N_DIM + n];
    float f = ETA_C * th + (1.f-ETA_C) * bow;
    if (n == ib) f = INFINITY;
    fuse[(size_t)brow*N_DIM + n] = f;
  }
}

// ---------------------------------------------------------------------------
// BatchNorm stats -> per-column affine (scale = g/sqrt(var+eps),
// shift = beta - mean*scale). Block owns 16 consecutive columns, row-major
// coalesced reads; deterministic LDS tree reduction.
// ---------------------------------------------------------------------------
__global__ __launch_bounds__(256) void bn_stats(
    const float* __restrict__ x, const float* __restrict__ g,
    const float* __restrict__ beta,
    float* __restrict__ scale, float* __restrict__ shift, int M, int N)
{
  __shared__ float sa[256], sb[256];
  const int tid     = threadIdx.x;
  const int colBase = blockIdx.x * 16;
  const int col     = colBase + (tid & 15);
  const int r0      = tid >> 4;          // 0..15
  float a = 0.f, b = 0.f;
  if (col < N) {
    for (int rI = r0; rI < M; rI += 16) {
      const float v = x[(size_t)rI*N + col];
      a += v; b += v*v;
    }
  }
  sa[tid] = a; sb[tid] = b; __syncthreads();
  for (int s = 128; s >= 16; s >>= 1) {
    if (tid < s) { sa[tid] += sa[tid+s]; sb[tid] += sb[tid+s]; }
    __syncthreads();
  }
  if (tid < 16 && colBase + tid < N) {
    const int c = colBase + tid;
    const float mean = sa[tid] / (float)M;
    const float var  = sb[tid] / (float)M - mean*mean;
    const float sc   = g[c] / sqrtf(var + EPS_BN);
    scale[c] = sc;
    shift[c] = beta[c] - mean * sc;
  }
}

// ---------------------------------------------------------------------------
// Per-row: BN-affine -> softmax over T -> VQ argmin vs codebook E ->
// quant row (= E[argmin]), s2 = sum(soft^2), qpart = sum((q - soft)^2).
// ---------------------------------------------------------------------------
__global__ __launch_bounds__(256) void softmax_quant(
    const float* __restrict__ theta_pre, const float* __restrict__ scale_t,
    const float* __restrict__ shift_t, const float* __restrict__ E,
    float* __restrict__ soft, float* __restrict__ quant,
    float* __restrict__ s2v, float* __restrict__ qpart)
{
  const int b = blockIdx.x * blockDim.x + threadIdx.x;
  if (b >= B_DIM) return;
  float z[T_DIM];
  float mx = -INFINITY;
  #pragma unroll
  for (int t = 0; t < T_DIM; ++t) {
    const float v = theta_pre[(size_t)b*T_DIM + t] * scale_t[t] + shift_t[t];
    z[t] = v; mx = fmaxf(mx, v);
  }
  float sum = 0.f;
  #pragma unroll
  for (int t = 0; t < T_DIM; ++t) { z[t] = expf(z[t] - mx); sum += z[t]; }
  const float inv = 1.f / sum;
  float ss = 0.f;
  #pragma unroll
  for (int t = 0; t < T_DIM; ++t) { z[t] *= inv; ss += z[t]*z[t]; }
  s2v[b] = ss;
  int best = 0; float bestd = INFINITY;
  for (int j = 0; j < T_DIM; ++j) {
    float d = 0.f;
    #pragma unroll
    for (int t = 0; t < T_DIM; ++t) { const float df = z[t] - E[j*T_DIM + t]; d += df*df; }
    if (d < bestd) { bestd = d; best = j; }
  }
  float qp = 0.f;
  #pragma unroll
  for (int t = 0; t < T_DIM; ++t) {
    const float q = E[best*T_DIM + t];
    quant[(size_t)b*T_DIM + t] = q;
    soft[(size_t)b*T_DIM + t]  = z[t];
    const float df = q - z[t]; qp += df*df;
  }
  qpart[b] = qp;
}

// ---------------------------------------------------------------------------
// Row-wise log-softmax of BN-affine(dec), in place: p[v] <- z[v] - lse(z)
// ---------------------------------------------------------------------------
__global__ __launch_bounds__(256) void logsoftmax_rows(
    float* __restrict__ p0, const float* __restrict__ scale,
    const float* __restrict__ shift, int N)
{
  __shared__ float red[256];
  const int row = blockIdx.x, tid = threadIdx.x;
  float* p = p0 + (size_t)row*N;
  float mx = -INFINITY;
  for (int v = tid; v < N; v += 256) mx = fmaxf(mx, p[v]*scale[v] + shift[v]);
  red[tid] = mx; __syncthreads();
  for (int s = 128; s > 0; s >>= 1) { if (tid < s) red[tid] = fmaxf(red[tid], red[tid+s]); __syncthreads(); }
  mx = red[0]; __syncthreads();
  float sum = 0.f;
  for (int v = tid; v < N; v += 256) sum += expf(p[v]*scale[v] + shift[v] - mx);
  red[tid] = sum; __syncthreads();
  for (int s = 128; s > 0; s >>= 1) { if (tid < s) red[tid] += red[tid+s]; __syncthreads(); }
  const float lse = mx + logf(red[0]);
  __syncthreads();
  for (int v = tid; v < N; v += 256) p[v] = p[v]*scale[v] + shift[v] - lse;
}

__global__ void tb2_kernel(const float* __restrict__ tb, float* __restrict__ tb2, int N)
{
  const int n = blockIdx.x * blockDim.x + threadIdx.x;
  if (n < N) {
    float s = 0.f;
    #pragma unroll
    for (int t = 0; t < T_DIM; ++t) { const float v = tb[(size_t)n*T_DIM + t]; s += v*v; }
    tb2[n] = s;
  }
}

// ---------------------------------------------------------------------------
// Top-20 smallest per fuse row: stage 10000-float row in LDS (40KB), then
// 20 deterministic block-argmin rounds (value, then index tie-break).
// ---------------------------------------------------------------------------
__global__ __launch_bounds__(256) void topk_kernel(
    const float* __restrict__ fuse, int* __restrict__ topk)
{
  __shared__ float srow[N_DIM];
  __shared__ float rv[256];
  __shared__ int   ri[256];
  const int row = blockIdx.x, tid = threadIdx.x;
  for (int n = tid; n < N_DIM; n += 256) srow[n] = fuse[(size_t)row*N_DIM + n];
  __syncthreads();
  for (int r = 0; r < K_SEL; ++r) {
    float bv = INFINITY; int bi = 0x7fffffff;
    for (int n = tid; n < N_DIM; n += 256) {
      const float v = srow[n];
      if (v < bv) { bv = v; bi = n; }
    }
    rv[tid] = bv; ri[tid] = bi; __syncthreads();
    for (int s = 128; s > 0; s >>= 1) {
      if (tid < s) {
        if (rv[tid+s] < rv[tid] || (rv[tid+s] == rv[tid] && ri[tid+s] < ri[tid])) {
          rv[tid] = rv[tid+s]; ri[tid] = ri[tid+s];
        }
      }
      __syncthreads();
    }
    if (tid == 0) { topk[row*K_SEL + r] = ri[0]; srow[ri[0]] = INFINITY; }
    __syncthreads();
  }
}

// ---------------------------------------------------------------------------
// Per-row cross-entropy with on-the-fly augmentation gather.
// ---------------------------------------------------------------------------
__global__ __launch_bounds__(256) void rec_loss_kernel(
    const float* __restrict__ inputs, const float* __restrict__ training,
    const int* __restrict__ topk, const float* __restrict__ logrecon,
    const int* __restrict__ is_aug, float* __restrict__ rowloss)
{
  __shared__ float red[256];
  const int row = blockIdx.x, tid = threadIdx.x;
  const int aug = is_aug[0];
  int tk[K_SEL];
  #pragma unroll
  for (int k = 0; k < K_SEL; ++k) tk[k] = topk[row*K_SEL + k];
  float acc = 0.f;
  if (aug) {
    for (int v = tid; v < V_DIM; v += 256) {
      float s = 0.f;
      #pragma unroll
      for (int k = 0; k < K_SEL; ++k) s += training[(size_t)tk[k]*V_DIM + v];
      const float x = inputs[(size_t)row*V_DIM + v] + (ALPHA_C/(float)K_SEL)*s;
      acc -= x * logrecon[(size_t)row*V_DIM + v];
    }
  } else {
    for (int v = tid; v < V_DIM; v += 256)
      acc -= inputs[(size_t)row*V_DIM + v] * logrecon[(size_t)row*V_DIM + v];
  }
  red[tid] = acc; __syncthreads();
  for (int s = 128; s > 0; s >>= 1) { if (tid < s) red[tid] += red[tid+s]; __syncthreads(); }
  if (tid == 0) rowloss[row] = red[0];
}

__global__ __launch_bounds__(256) void final_kernel(
    const float* __restrict__ rowloss, const float* __restrict__ qpart,
    float* __restrict__ out)
{
  __shared__ float r1[256], r2[256];
  const int tid = threadIdx.x;
  float a = 0.f, b = 0.f;
  for (int i = tid; i < B_DIM; i += 256) { a += rowloss[i]; b += qpart[i]; }
  r1[tid] = a; r2[tid] = b; __syncthreads();
  for (int s = 128; s > 0; s >>= 1) {
    if (tid < s) { r1[tid] += r1[tid+s]; r2[tid] += r2[tid+s]; }
    __syncthreads();
  }
  if (tid == 0)
    out[0] = r1[0] / (float)B_DIM + (1.f + CC_C) * r2[0] / (float)(B_DIM * T_DIM);
}

// ---------------------------------------------------------------------------
extern "C" void kernel_launch(void* const* d_in, const int* in_sizes, int n_in,
                              void* d_out, int out_size, void* d_ws, size_t ws_size,
                              hipStream_t stream)
{
  const int*   idx       = (const int*)d_in[0];
  const float* inputs    = (const float*)d_in[1];
  const int*   is_aug    = (const int*)d_in[2];
  const float* training  = (const float*)d_in[3];
  const float* Mcos      = (const float*)d_in[4];
  const float* Mcoo      = (const float*)d_in[5];
  const float* tbank     = (const float*)d_in[6];
  const float* W11       = (const float*)d_in[7];
  const float* b11       = (const float*)d_in[8];
  const float* W12       = (const float*)d_in[9];
  const float* b12       = (const float*)d_in[10];
  const float* W21       = (const float*)d_in[11];
  const float* b21       = (const float*)d_in[12];
  const float* g_mean    = (const float*)d_in[13];
  const float* beta_mean = (const float*)d_in[14];
  const float* g_dec     = (const float*)d_in[15];
  const float* beta_dec  = (const float*)d_in[16];
  const float* Wd        = (const float*)d_in[17];
  const float* E         = (const float*)d_in[18];
  float* out = (float*)d_out;

  char* w = (char*)d_ws;
  size_t off = 0;
  auto alloc = [&](size_t bytes) -> char* {
    char* p = w + off; off += (bytes + 255) & ~(size_t)255; return p;
  };
  float* e1a       = (float*)alloc(sizeof(float) * B_DIM * H_DIM);
  float* e1b       = (float*)alloc(sizeof(float) * B_DIM * H_DIM);
  float* theta_pre = (float*)alloc(sizeof(float) * B_DIM * T_DIM);
  float* soft      = (float*)alloc(sizeof(float) * B_DIM * T_DIM);
  float* quant     = (float*)alloc(sizeof(float) * B_DIM * T_DIM);
  float* s2v       = (float*)alloc(sizeof(float) * B_DIM);
  float* qpart     = (float*)alloc(sizeof(float) * B_DIM);
  float* rowloss   = (float*)alloc(sizeof(float) * B_DIM);
  float* scale_t   = (float*)alloc(sizeof(float) * T_DIM);
  float* shift_t   = (float*)alloc(sizeof(float) * T_DIM);
  float* scale_v   = (float*)alloc(sizeof(float) * V_DIM);
  float* shift_v   = (float*)alloc(sizeof(float) * V_DIM);
  float* tb2v      = (float*)alloc(sizeof(float) * N_DIM);
  float* dec       = (float*)alloc(sizeof(float) * (size_t)B_DIM * V_DIM); // -> logrecon
  float* fuse      = (float*)alloc(sizeof(float) * (size_t)B_DIM * N_DIM);
  int*   topk      = (int*)  alloc(sizeof(int)   * B_DIM * K_SEL);

  const dim3 wave(32);

  // encoder
  wmma_gemm_nt<<<dim3(B_DIM/16, (H_DIM+15)/16), wave, 0, stream>>>(inputs, W11, b11, e1a, B_DIM, H_DIM, V_DIM, 1);
  wmma_gemm_nt<<<dim3(B_DIM/16, (H_DIM+15)/16), wave, 0, stream>>>(e1a, W12, b12, e1b, B_DIM, H_DIM, H_DIM, 1);
  wmma_gemm_nt<<<dim3(B_DIM/16, (T_DIM+15)/16), wave, 0, stream>>>(e1b, W21, b21, theta_pre, B_DIM, T_DIM, H_DIM, 0);

  // BN(mean-path) + softmax + VQ
  bn_stats<<<dim3((T_DIM+15)/16), dim3(256), 0, stream>>>(theta_pre, g_mean, beta_mean, scale_t, shift_t, B_DIM, T_DIM);
  softmax_quant<<<dim3(B_DIM/256), dim3(256), 0, stream>>>(theta_pre, scale_t, shift_t, E, soft, quant, s2v, qpart);

  // decoder + BN(dec) + log-softmax (in place)
  wmma_gemm_nt<<<dim3(B_DIM/16, (V_DIM+15)/16), wave, 0, stream>>>(quant, Wd, (const float*)nullptr, dec, B_DIM, V_DIM, T_DIM, 0);
  bn_stats<<<dim3((V_DIM+15)/16), dim3(256), 0, stream>>>(dec, g_dec, beta_dec, scale_v, shift_v, B_DIM, V_DIM);
  logsoftmax_rows<<<dim3(B_DIM), dim3(256), 0, stream>>>(dec, scale_v, shift_v, V_DIM);

  // augmentation: fuse matrix + top-K
  tb2_kernel<<<dim3((N_DIM+255)/256), dim3(256), 0, stream>>>(tbank, tb2v, N_DIM);
  wmma_cost<<<dim3(B_DIM/16, N_DIM/16), wave, 0, stream>>>(soft, tbank, s2v, tb2v, idx, Mcos, Mcoo, fuse);
  topk_kernel<<<dim3(B_DIM), dim3(256), 0, stream>>>(fuse, topk);

  // losses
  rec_loss_kernel<<<dim3(B_DIM), dim3(256), 0, stream>>>(inputs, training, topk, dec, is_aug, rowloss);
  final_kernel<<<dim3(1), dim3(256), 0, stream>>>(rowloss, qpart, out);
}